// TensorCircuit_8770323218960
// MI455X (gfx1250) — compile-verified
//
#include <hip/hip_runtime.h>

#define NUM_VARS 64
#define KDIM 32
#define NUM_CATS 256
#define NUM_INPUT (NUM_VARS * KDIM)   // 2048
#define LLAYERS 4
#define EDIM 16384
#define NDIM 8192
#define CDIM 4
#define BDIM 1024
#define TOTAL_ROWS (NUM_INPUT + LLAYERS * NDIM)  // 34816
#define NCHUNKS 32                    // n-chunks for root reduction (256 n each)

typedef __attribute__((ext_vector_type(2))) float v2f;
typedef __attribute__((ext_vector_type(8))) float v8f;

// --- 1. transpose inputs (B x V int32) -> xT (V x B) for coalesced reuse ---
__global__ void k_transpose_inputs(const int* __restrict__ inputs, int* __restrict__ xT) {
    int idx = blockIdx.x * 256 + threadIdx.x;     // 65536 threads
    int b = idx >> 6;                             // coalesced read of inputs
    int v = idx & (NUM_VARS - 1);
    xT[v * BDIM + b] = inputs[b * NUM_VARS + v];
}

// --- 2. input layer gather: nm[r, b] = input_logp[r, xT[v, b]] ---
__global__ void k_input_gather(const float* __restrict__ logp, const int* __restrict__ xT,
                               float* __restrict__ nm) {
    int idx = blockIdx.x * 256 + threadIdx.x;     // 2048 * 1024 threads
    int b = idx & (BDIM - 1);
    int r = idx >> 10;                            // r = v*K + k
    int v = r >> 5;                               // r / KDIM
    int cat = xT[v * BDIM + b];
    nm[idx] = logp[r * NUM_CATS + cat];           // gather within 1KB row (cache-resident)
}

// --- 3. fused product+sum layer: out[n,b] = LSE_c( logw[n,c] + nm[p0_c,b] + nm[p1_c,b] ) ---
__global__ void __launch_bounds__(256)
k_layer_sum(const float* __restrict__ nm, float* __restrict__ out,
            const int* __restrict__ prod, const int* __restrict__ sch,
            const float* __restrict__ slw) {
    int n = blockIdx.y;                           // sum node
    int b = blockIdx.x * 256 + threadIdx.x;       // batch column
    float t[CDIM];
#pragma unroll
    for (int c = 0; c < CDIM; ++c) {
        int e  = sch[n * CDIM + c];               // child product node (uniform per block)
        int a0 = prod[e * 2 + 0];
        int a1 = prod[e * 2 + 1];
        t[c] = slw[n * CDIM + c] + nm[(size_t)a0 * BDIM + b] + nm[(size_t)a1 * BDIM + b];
    }
    float m = fmaxf(fmaxf(t[0], t[1]), fmaxf(t[2], t[3]));
    float s = __expf(t[0] - m) + __expf(t[1] - m) + __expf(t[2] - m) + __expf(t[3] - m);
    out[(size_t)n * BDIM + b] = m + __logf(s);
}

// --- 4a. root: partial column max over 256-row n-chunks ---
__global__ void k_root_max_part(const float* __restrict__ last, const float* __restrict__ rlw,
                                float* __restrict__ pmax) {
    int b  = blockIdx.x * 256 + threadIdx.x;
    int ny = blockIdx.y;
    float m = -__builtin_inff();
    for (int i = 0; i < NDIM / NCHUNKS; ++i) {
        int n = ny * (NDIM / NCHUNKS) + i;
        m = fmaxf(m, rlw[n] + last[(size_t)n * BDIM + b]);   // coalesced row reads
    }
    pmax[ny * BDIM + b] = m;
}

// --- 4b. combine partial maxes ---
__global__ void k_root_max_comb(const float* __restrict__ pmax, float* __restrict__ mcol) {
    int b = blockIdx.x * 256 + threadIdx.x;
    float v = pmax[b];
    for (int j = 1; j < NCHUNKS; ++j) v = fmaxf(v, pmax[j * BDIM + b]);
    mcol[b] = v;
}

// --- 4c. WMMA mat-vec: psum[chunk,b] = sum_{n in chunk} exp(rlw[n]) * exp(last[n,b]-m[b]) ---
// One wave per block. A(16x4) = weight row replicated over M; B(4x16) = exp'd tile.
// A layout (ISA 7.12.2, 32-bit 16x4): lanes 0-15 -> K=0,1 in v0,v1; lanes 16-31 -> K=2,3.
// B layout (4x16, 2 VGPRs): lanes 0-15 -> K=0(v0)/K=1(v1), lanes 16-31 -> K=2/K=3, N=lane%16.
// D rows are all identical; row M=0 lives in acc VGPR0, lanes 0-15.
__global__ void __launch_bounds__(32)
k_root_wmma_sum(const float* __restrict__ last, const float* __restrict__ rlw,
                const float* __restrict__ mcol, float* __restrict__ psum) {
    int lane  = threadIdx.x;                     // 0..31
    int b0    = blockIdx.x * 16;                 // 64 column tiles
    int n0    = blockIdx.y * (NDIM / NCHUNKS);   // 32 row chunks of 256
    int col   = b0 + (lane & 15);
    int khalf = (lane >> 4) * 2;                 // 0 for lanes 0-15, 2 for lanes 16-31
    float mneg = -mcol[col];

    v8f acc = {};
    for (int k0 = n0; k0 < n0 + NDIM / NCHUNKS; k0 += 4) {
        v2f a, bv;
        a.x  = __expf(rlw[k0 + khalf]);
        a.y  = __expf(rlw[k0 + khalf + 1]);
        bv.x = __expf(last[(size_t)(k0 + khalf)     * BDIM + col] + mneg);
        bv.y = __expf(last[(size_t)(k0 + khalf + 1) * BDIM + col] + mneg);
        acc = __builtin_amdgcn_wmma_f32_16x16x4_f32(
            /*neg_a=*/false, a, /*neg_b=*/false, bv,
            /*c_mod=*/(short)0, acc, /*reuse_a=*/false, /*reuse_b=*/false);
    }
    if (lane < 16) psum[blockIdx.y * BDIM + col] = acc[0];   // D[M=0][N=lane]
}

// --- 4d. final: out[b] = m[b] + log( sum_chunks psum ) ---
__global__ void k_root_final(const float* __restrict__ psum, const float* __restrict__ mcol,
                             float* __restrict__ out) {
    int b = blockIdx.x * 256 + threadIdx.x;
    float s = 0.0f;
    for (int j = 0; j < NCHUNKS; ++j) s += psum[j * BDIM + b];
    out[b] = mcol[b] + __logf(s);
}

extern "C" void kernel_launch(void* const* d_in, const int* in_sizes, int n_in,
                              void* d_out, int out_size, void* d_ws, size_t ws_size,
                              hipStream_t stream) {
    (void)in_sizes; (void)n_in; (void)out_size; (void)ws_size;
    const int*   inputs     = (const int*)d_in[0];   // (B, NUM_VARS) int32
    const float* input_logp = (const float*)d_in[1]; // (V, K, 256) f32
    const int*   prod_ids   = (const int*)d_in[2];   // (L, E, 2) int32
    const int*   sum_ch_ids = (const int*)d_in[3];   // (L, N, C) int32
    const float* sum_logw   = (const float*)d_in[4]; // (L, N, C) f32
    const float* root_logw  = (const float*)d_in[5]; // (N,) f32
    float* out = (float*)d_out;                      // (B, 1) f32

    // workspace layout
    float* nm = (float*)d_ws;                                    // 34816 x 1024 f32
    size_t nm_elems = (size_t)TOTAL_ROWS * BDIM;
    int*   xT   = (int*)(nm + nm_elems);                         // 64 x 1024 int32
    float* pmax = (float*)(xT + NUM_VARS * BDIM);                // 32 x 1024 f32
    float* mcol = pmax + NCHUNKS * BDIM;                         // 1024 f32
    float* psum = mcol + BDIM;                                   // 32 x 1024 f32

    k_transpose_inputs<<<(NUM_VARS * BDIM) / 256, 256, 0, stream>>>(inputs, xT);
    k_input_gather<<<(NUM_INPUT * BDIM) / 256, 256, 0, stream>>>(input_logp, xT, nm);

    for (int l = 0; l < LLAYERS; ++l) {
        k_layer_sum<<<dim3(BDIM / 256, NDIM), 256, 0, stream>>>(
            nm, nm + (size_t)(NUM_INPUT + l * NDIM) * BDIM,
            prod_ids + (size_t)l * EDIM * 2,
            sum_ch_ids + (size_t)l * NDIM * CDIM,
            sum_logw + (size_t)l * NDIM * CDIM);
    }

    const float* last = nm + (size_t)(NUM_INPUT + (LLAYERS - 1) * NDIM) * BDIM;
    k_root_max_part<<<dim3(BDIM / 256, NCHUNKS), 256, 0, stream>>>(last, root_logw, pmax);
    k_root_max_comb<<<BDIM / 256, 256, 0, stream>>>(pmax, mcol);
    k_root_wmma_sum<<<dim3(BDIM / 16, NCHUNKS), 32, 0, stream>>>(last, root_logw, mcol, psum);
    k_root_final<<<BDIM / 256, 256, 0, stream>>>(psum, mcol, out);
}